// PraxisGraph_88742614270013
// MI455X (gfx1250) — compile-verified
//
#include <hip/hip_runtime.h>
#include <hip/hip_bf16.h>

#define Hdim 1024
#define Bsz  16
#define Ssz  2048
#define Lsz  32
#define EDGE 256
#define SPLIT 32
#define ROWS_PER_BLOCK (Ssz / SPLIT)   /* 64 rows per block, 8 per wave */
#define LN_EPS 1e-5f

typedef __attribute__((ext_vector_type(2))) float v2f;
typedef __attribute__((ext_vector_type(8))) float v8f;

// ---------------------------------------------------------------------------
// Kernel 1: per-row LayerNorm (pre-gamma/beta) + accumulate sum over rows.
// One wave per row; wave32 shuffle reduction for mean/var; LDS combine.
// partial[b*SPLIT + split][h] = sum over 64 rows of (x - mu) * rsqrt(var+eps)
// ---------------------------------------------------------------------------
__global__ void __launch_bounds__(256) ln_partial_kernel(
    const float* __restrict__ hs, float* __restrict__ partial) {
  __shared__ float red[8][Hdim];                 // 32 KB
  const int blk   = blockIdx.x;                  // b*SPLIT + split
  const int b     = blk / SPLIT;
  const int split = blk % SPLIT;
  const int wave  = threadIdx.x >> 5;            // 0..7
  const int lane  = threadIdx.x & 31;
  const float* base = hs + (size_t)(b * Ssz + split * ROWS_PER_BLOCK) * Hdim;

  float4 acc[8];
#pragma unroll
  for (int i = 0; i < 8; ++i) acc[i] = make_float4(0.f, 0.f, 0.f, 0.f);

  for (int r = wave; r < ROWS_PER_BLOCK; r += 8) {
    const float* row = base + (size_t)r * Hdim;
    if (r + 8 < ROWS_PER_BLOCK)
      __builtin_prefetch(row + 8 * Hdim + lane * 4, 0, 1);  // global_prefetch_b8

    float4 x[8];
    float s = 0.f, s2 = 0.f;
#pragma unroll
    for (int i = 0; i < 8; ++i) {
      x[i] = *(const float4*)(row + i * 128 + lane * 4);
      s  += x[i].x + x[i].y + x[i].z + x[i].w;
      s2 += x[i].x * x[i].x + x[i].y * x[i].y + x[i].z * x[i].z + x[i].w * x[i].w;
    }
#pragma unroll
    for (int off = 16; off > 0; off >>= 1) {     // wave32 reduction
      s  += __shfl_xor(s,  off, 32);
      s2 += __shfl_xor(s2, off, 32);
    }
    const float mu  = s * (1.0f / Hdim);
    const float var = s2 * (1.0f / Hdim) - mu * mu;
    const float rs  = rsqrtf(var + LN_EPS);
#pragma unroll
    for (int i = 0; i < 8; ++i) {
      acc[i].x += (x[i].x - mu) * rs;
      acc[i].y += (x[i].y - mu) * rs;
      acc[i].z += (x[i].z - mu) * rs;
      acc[i].w += (x[i].w - mu) * rs;
    }
  }

  // deterministic cross-wave combine through LDS
#pragma unroll
  for (int i = 0; i < 8; ++i)
    *(float4*)&red[wave][i * 128 + lane * 4] = acc[i];
  __syncthreads();

  const int h = threadIdx.x * 4;                 // 256 threads * 4 = 1024
  float4 tot = make_float4(0.f, 0.f, 0.f, 0.f);
#pragma unroll
  for (int w = 0; w < 8; ++w) {
    float4 p = *(const float4*)&red[w][h];
    tot.x += p.x; tot.y += p.y; tot.z += p.z; tot.w += p.w;
  }
  *(float4*)(partial + (size_t)blk * Hdim + h) = tot;
}

// ---------------------------------------------------------------------------
// Kernel 2 (48 blocks):
//   blocks [0,16):  query_input[b] = layer_emb[cl] + g*mean + b
//   blocks [16,48): layer_features[l] = layer_emb[l] + cent_emb[l]
// Precomputing layer_features keeps the WMMA kernel's inner loop branch-free.
// ---------------------------------------------------------------------------
__global__ void __launch_bounds__(256) finalize_kernel(
    const float* __restrict__ partial, const float* __restrict__ le,
    const float* __restrict__ ce, const float* __restrict__ g,
    const float* __restrict__ bt, const int* __restrict__ cur,
    float* __restrict__ qi, float* __restrict__ lf) {
  const int blk = blockIdx.x;
  const int h = threadIdx.x * 4;
  if (blk < Bsz) {
    const int b = blk;
    float4 s = make_float4(0.f, 0.f, 0.f, 0.f);
#pragma unroll 4
    for (int sp = 0; sp < SPLIT; ++sp) {
      float4 p = *(const float4*)(partial + (size_t)(b * SPLIT + sp) * Hdim + h);
      s.x += p.x; s.y += p.y; s.z += p.z; s.w += p.w;
    }
    const int cl = cur[0];
    const float inv = 1.0f / (float)Ssz;
    float4 gv = *(const float4*)(g + h);
    float4 bv = *(const float4*)(bt + h);
    float4 lv = *(const float4*)(le + (size_t)cl * Hdim + h);
    float4 o;
    o.x = lv.x + gv.x * s.x * inv + bv.x;
    o.y = lv.y + gv.y * s.y * inv + bv.y;
    o.z = lv.z + gv.z * s.z * inv + bv.z;
    o.w = lv.w + gv.w * s.w * inv + bv.w;
    *(float4*)(qi + (size_t)b * Hdim + h) = o;
  } else {
    const int l = blk - Bsz;
    float4 a = *(const float4*)(le + (size_t)l * Hdim + h);
    float4 c = *(const float4*)(ce + (size_t)l * Hdim + h);
    float4 o = make_float4(a.x + c.x, a.y + c.y, a.z + c.z, a.w + c.w);
    *(float4*)(lf + (size_t)l * Hdim + h) = o;
  }
}

// ---------------------------------------------------------------------------
// Kernel 3: q/k projection via V_WMMA_F32_16X16X4_F32.
// 192 waves: waves [0,64)  -> q tile (A = query_input [16xH],  W = Wq)
//            waves [64,192)-> k tiles (A = layer_features,     W = Wk), M=32
// A layout (ISA 7.12.2, 32-bit A 16x4): lanes 0-15 M=lane; VGPR0 holds K=0
// (lanes 0-15) / K=2 (lanes 16-31); VGPR1 holds K=1 / K=3.
// B (4x16) mirrors with N across lanes. D: VGPR j -> M=j (+8 for lanes>=16).
// Inner loop is branch-free: 2x global_load_b64 -> v_wmma per K-step,
// 4 interleaved accumulators to cover XDL latency.
// ---------------------------------------------------------------------------
__global__ void __launch_bounds__(256) proj_kernel(
    const float* __restrict__ qi, const float* __restrict__ lf,
    const float* __restrict__ Wq, const float* __restrict__ bq,
    const float* __restrict__ Wk, const float* __restrict__ bk,
    float* __restrict__ qout, float* __restrict__ kout) {
  const int w    = blockIdx.x * 8 + (threadIdx.x >> 5);  // 0..191
  const int lane = threadIdx.x & 31;
  const int half = lane >> 4;
  const int m16  = lane & 15;                  // M row for A, N col for B/D

  const bool is_q = (w < 64);
  const int  t    = is_q ? w : (w - 64);
  const int  mt   = is_q ? 0 : (t >> 6);       // k M-tile: 0 or 1
  const int  n0   = (is_q ? t : (t & 63)) * 16;

  const float* W    = is_q ? Wq : Wk;
  const float* bias = is_q ? bq : bk;
  const float* a0   = is_q ? (qi + (size_t)m16 * Hdim)
                           : (lf + (size_t)(mt * 16 + m16) * Hdim);
  const float* wrow = W + (size_t)(n0 + m16) * Hdim;

  v8f cacc[4];
  const v8f zero = {0.f, 0.f, 0.f, 0.f, 0.f, 0.f, 0.f, 0.f};
#pragma unroll
  for (int u = 0; u < 4; ++u) cacc[u] = zero;

  for (int k0 = 0; k0 < Hdim; k0 += 16) {
#pragma unroll
    for (int u = 0; u < 4; ++u) {
      const int idx = k0 + u * 4 + 2 * half;
      float2 p  = *(const float2*)(a0 + idx);
      float2 bw = *(const float2*)(wrow + idx);
      v2f av, bv;
      av[0] = p.x;  av[1] = p.y;
      bv[0] = bw.x; bv[1] = bw.y;
      cacc[u] = __builtin_amdgcn_wmma_f32_16x16x4_f32(
          false, av, false, bv, (short)0, cacc[u], false, false);
    }
  }
  v8f c = cacc[0] + cacc[1] + cacc[2] + cacc[3];

  float* out = is_q ? qout : kout;
  const float bb = bias[n0 + m16];
#pragma unroll
  for (int j = 0; j < 8; ++j) {
    const int m = j + 8 * half + (is_q ? 0 : mt * 16);
    out[(size_t)m * Hdim + n0 + m16] = c[j] + bb;
  }
}

// ---------------------------------------------------------------------------
// Kernel 4: edge bias + attention + spatial bias + wave32 softmax over L=32.
// One block, 16 waves; wave b handles batch b, lane l handles layer l.
// ---------------------------------------------------------------------------
__global__ void __launch_bounds__(512) scores_kernel(
    const float* __restrict__ q, const float* __restrict__ k,
    const float* __restrict__ edge, const float* __restrict__ spatial,
    const int* __restrict__ avail, const int* __restrict__ cur,
    float* __restrict__ out) {
  __shared__ float epart[16][Lsz];
  __shared__ float ebias[Lsz];
  const int tid  = threadIdx.x;
  const int w    = tid >> 5;
  const int lane = tid & 31;
  const int cl   = cur[0];

  // edge_bias[j] = sum_i valid(i) * dot(E[i],E[j]) / max(|i-cl|,1)
  float contrib = 0.f;
  const float* El = edge + (size_t)lane * EDGE;
#pragma unroll
  for (int rep = 0; rep < 2; ++rep) {
    const int ii = w + rep * 16;
    const bool valid = (avail[ii] > 0) && (ii != cl);   // wave-uniform
    if (valid) {
      const float* Ei = edge + (size_t)ii * EDGE;
      float d = 0.f;
      for (int t2 = 0; t2 < EDGE; t2 += 4) {
        float4 a  = *(const float4*)(Ei + t2);
        float4 b2 = *(const float4*)(El + t2);
        d += a.x * b2.x + a.y * b2.y + a.z * b2.z + a.w * b2.w;
      }
      const int di = (ii > cl) ? (ii - cl) : (cl - ii);
      contrib += d / (float)(di > 0 ? di : 1);
    }
  }
  epart[w][lane] = contrib;
  __syncthreads();
  if (tid < Lsz) {
    float e = 0.f;
#pragma unroll
    for (int w2 = 0; w2 < 16; ++w2) e += epart[w2][tid];
    ebias[tid] = e;
  }
  __syncthreads();

  // attention row for batch b = w
  const float* qr = q + (size_t)w * Hdim;
  const float* kr = k + (size_t)lane * Hdim;
  float att = 0.f;
  for (int t2 = 0; t2 < Hdim; t2 += 4) {
    float4 qq = *(const float4*)(qr + t2);
    float4 kk = *(const float4*)(kr + t2);
    att += qq.x * kk.x + qq.y * kk.y + qq.z * kk.z + qq.w * kk.w;
  }
  const int dist = (lane > cl) ? (lane - cl) : (cl - lane);
  float score = att * (1.0f / 32.0f) + spatial[dist] + ebias[lane];

  // softmax across the wave (L == 32 == wave32)
  float m = score;
#pragma unroll
  for (int off = 16; off > 0; off >>= 1) m = fmaxf(m, __shfl_xor(m, off, 32));
  const float e = __expf(score - m);
  float sum = e;
#pragma unroll
  for (int off = 16; off > 0; off >>= 1) sum += __shfl_xor(sum, off, 32);
  out[w * Lsz + lane] = e / sum;
}

// ---------------------------------------------------------------------------
extern "C" void kernel_launch(void* const* d_in, const int* in_sizes, int n_in,
                              void* d_out, int out_size, void* d_ws, size_t ws_size,
                              hipStream_t stream) {
  const float* hs    = (const float*)d_in[0];
  const float* le    = (const float*)d_in[1];
  const float* ce    = (const float*)d_in[2];
  const float* sp    = (const float*)d_in[3];
  const float* ee    = (const float*)d_in[4];
  const float* g     = (const float*)d_in[5];
  const float* bt    = (const float*)d_in[6];
  const float* Wq    = (const float*)d_in[7];
  const float* bq    = (const float*)d_in[8];
  const float* Wk    = (const float*)d_in[9];
  const float* bk    = (const float*)d_in[10];
  const int*   avail = (const int*)d_in[11];
  const int*   cl    = (const int*)d_in[12];

  float* ws      = (float*)d_ws;
  float* partial = ws;                                   // [B*SPLIT][H]  2 MB
  float* qi      = partial + (size_t)Bsz * SPLIT * Hdim; // [B][H]
  float* lf      = qi + (size_t)Bsz * Hdim;              // [L][H]
  float* qo      = lf + (size_t)Lsz * Hdim;              // [B][H]
  float* ko      = qo + (size_t)Bsz * Hdim;              // [L][H]
  float* out     = (float*)d_out;

  ln_partial_kernel<<<Bsz * SPLIT, 256, 0, stream>>>(hs, partial);
  finalize_kernel<<<Bsz + Lsz, 256, 0, stream>>>(partial, le, ce, g, bt, cl, qi, lf);
  proj_kernel<<<24, 256, 0, stream>>>(qi, lf, Wq, bq, Wk, bk, qo, ko);
  scores_kernel<<<1, 512, 0, stream>>>(qo, ko, ee, sp, avail, cl, out);
}